// QT_62130996904274
// MI455X (gfx1250) — compile-verified
//
#include <hip/hip_runtime.h>
#include <math.h>
#include <stdint.h>

// ---------------------------------------------------------------------------
// Quadtree decomposition, B=16, H=W=512, LEVEL=8, THRESH=0.2
// Workspace layout: float2 (sum, sumsq) per block, levels 0..8 concatenated.
// off[d] = 16 * (4^d - 1) / 3   (in float2 elements)
// total = 1,398,096 float2 = ~10.7 MB
// ---------------------------------------------------------------------------

__constant__ int kOff[9] = {0, 16, 80, 336, 1360, 5456, 21840, 87376, 349520};

// ---- CDNA5 async global->LDS copy (ASYNCcnt path) -------------------------
typedef int v4i __attribute__((ext_vector_type(4)));
typedef __attribute__((address_space(1))) v4i gv4i;
typedef __attribute__((address_space(3))) v4i lv4i;

__device__ __forceinline__ void async_ld_b128(const float* g, float* l) {
#if __has_builtin(__builtin_amdgcn_global_load_async_to_lds_b128)
  __builtin_amdgcn_global_load_async_to_lds_b128((gv4i*)g, (lv4i*)l, 0, 0);
#else
  unsigned loff =
      (unsigned)(unsigned long long)(__attribute__((address_space(3))) void*)l;
  asm volatile("global_load_async_to_lds_b128 %0, %1, off"
               :
               : "v"(loff), "v"(g)
               : "memory");
#endif
}

__device__ __forceinline__ void wait_async0() {
#if __has_builtin(__builtin_amdgcn_s_wait_asynccnt)
  __builtin_amdgcn_s_wait_asynccnt(0);
#else
  asm volatile("s_wait_asynccnt 0" ::: "memory");
#endif
}

// ---------------------------------------------------------------------------
// Phase 1: per 64x64 tile, build stats for levels 8..3.
// grid = 16 images * 64 tiles, block = 256 threads (8 waves).
// ---------------------------------------------------------------------------
__global__ __launch_bounds__(256) void qt_phase1(const float* __restrict__ x,
                                                 float2* __restrict__ ws) {
  __shared__ float tile[64 * 64];  // 16 KB; reused as reduction buffer later

  const int tid = threadIdx.x;
  const int b   = blockIdx.x >> 6;
  const int t   = blockIdx.x & 63;
  const int ty  = t >> 3, tx = t & 7;
  const float* img = x + ((size_t)b << 18);

  // Stage the 64x64 tile into LDS: 1024 x 16B chunks, 4 per thread,
  // consecutive lanes -> consecutive 16B (fully coalesced).
  for (int k = 0; k < 4; ++k) {
    int c  = (k << 8) + tid;   // chunk id 0..1023
    int r  = c >> 4;           // tile row 0..63
    int c4 = (c & 15) << 2;    // tile col (floats) 0..60
    const float* g = img + (size_t)(ty * 64 + r) * 512 + (tx * 64 + c4);
    async_ld_b128(g, &tile[r * 64 + c4]);
  }
  wait_async0();
  __syncthreads();

  // Each thread owns a 4x4 patch: (py,px) in 16x16 grid of patches.
  const int py = tid >> 4, px = tid & 15;
  const float4 v0 = *(const float4*)&tile[(py * 4 + 0) * 64 + px * 4];
  const float4 v1 = *(const float4*)&tile[(py * 4 + 1) * 64 + px * 4];
  const float4 v2 = *(const float4*)&tile[(py * 4 + 2) * 64 + px * 4];
  const float4 v3 = *(const float4*)&tile[(py * 4 + 3) * 64 + px * 4];

  // Level-8 stats: four 2x2 quadrants of the patch.
  float s00 = v0.x + v0.y + v1.x + v1.y;
  float q00 = v0.x * v0.x + v0.y * v0.y + v1.x * v1.x + v1.y * v1.y;
  float s01 = v0.z + v0.w + v1.z + v1.w;
  float q01 = v0.z * v0.z + v0.w * v0.w + v1.z * v1.z + v1.w * v1.w;
  float s10 = v2.x + v2.y + v3.x + v3.y;
  float q10 = v2.x * v2.x + v2.y * v2.y + v3.x * v3.x + v3.y * v3.y;
  float s11 = v2.z + v2.w + v3.z + v3.w;
  float q11 = v2.z * v2.z + v2.w * v2.w + v3.z * v3.z + v3.w * v3.w;

  // Level-8 writes: 2 blocks per float4 store (coalesced).
  {
    float2* ws8  = ws + 349520 + ((size_t)b << 16);  // 256x256 blocks
    float4* ws8v = (float4*)ws8;                     // 128 float4 per row
    int r8 = ty * 32 + py * 2;
    int c8 = tx * 16 + px;
    ws8v[(size_t)r8 * 128 + c8]       = make_float4(s00, q00, s01, q01);
    ws8v[(size_t)(r8 + 1) * 128 + c8] = make_float4(s10, q10, s11, q11);
  }

  // Level-7 (4x4 block == this patch).
  float s7 = s00 + s01 + s10 + s11;
  float q7 = q00 + q01 + q10 + q11;
  {
    float2* ws7 = ws + 87376 + ((size_t)b << 14);  // 128x128 blocks
    ws7[(size_t)(ty * 16 + py) * 128 + (tx * 16 + px)] = make_float2(s7, q7);
  }

  // Reuse tile LDS as float2 reduction buffer.
  __syncthreads();  // everyone done reading tile
  float2* red = (float2*)tile;
  red[tid] = make_float2(s7, q7);
  __syncthreads();

  // Level-6: 8x8 per tile.
  if (tid < 64) {
    int y = tid >> 3, xq = tid & 7;
    float2 a = red[(2 * y) * 16 + 2 * xq];
    float2 c = red[(2 * y) * 16 + 2 * xq + 1];
    float2 e = red[(2 * y + 1) * 16 + 2 * xq];
    float2 f = red[(2 * y + 1) * 16 + 2 * xq + 1];
    float2 r = make_float2(a.x + c.x + e.x + f.x, a.y + c.y + e.y + f.y);
    (ws + 21840 + ((size_t)b << 12))[(size_t)(ty * 8 + y) * 64 + (tx * 8 + xq)] = r;
    red[256 + tid] = r;
  }
  __syncthreads();

  // Level-5: 4x4 per tile.
  if (tid < 16) {
    int y = tid >> 2, xq = tid & 3;
    float2 a = red[256 + (2 * y) * 8 + 2 * xq];
    float2 c = red[256 + (2 * y) * 8 + 2 * xq + 1];
    float2 e = red[256 + (2 * y + 1) * 8 + 2 * xq];
    float2 f = red[256 + (2 * y + 1) * 8 + 2 * xq + 1];
    float2 r = make_float2(a.x + c.x + e.x + f.x, a.y + c.y + e.y + f.y);
    (ws + 5456 + ((size_t)b << 10))[(size_t)(ty * 4 + y) * 32 + (tx * 4 + xq)] = r;
    red[320 + tid] = r;
  }
  __syncthreads();

  // Level-4: 2x2 per tile.
  if (tid < 4) {
    int y = tid >> 1, xq = tid & 1;
    float2 a = red[320 + (2 * y) * 4 + 2 * xq];
    float2 c = red[320 + (2 * y) * 4 + 2 * xq + 1];
    float2 e = red[320 + (2 * y + 1) * 4 + 2 * xq];
    float2 f = red[320 + (2 * y + 1) * 4 + 2 * xq + 1];
    float2 r = make_float2(a.x + c.x + e.x + f.x, a.y + c.y + e.y + f.y);
    (ws + 1360 + ((size_t)b << 8))[(size_t)(ty * 2 + y) * 16 + (tx * 2 + xq)] = r;
    red[336 + tid] = r;
  }
  __syncthreads();

  // Level-3: 1 per tile.
  if (tid == 0) {
    float2 a = red[336], c = red[337], e = red[338], f = red[339];
    float2 r = make_float2(a.x + c.x + e.x + f.x, a.y + c.y + e.y + f.y);
    (ws + 336 + ((size_t)b << 6))[ty * 8 + tx] = r;
  }
}

// ---------------------------------------------------------------------------
// Phase 2: levels 3 -> 2 -> 1 -> 0. grid = 16 images, block = 64 threads.
// ---------------------------------------------------------------------------
__global__ __launch_bounds__(64) void qt_phase2(float2* __restrict__ ws) {
  __shared__ float2 red[96];
  const int b = blockIdx.x, tid = threadIdx.x;

  red[tid] = ws[336 + b * 64 + tid];  // level-3: 8x8 blocks
  __syncthreads();

  if (tid < 16) {  // level-2: 4x4
    int y = tid >> 2, xq = tid & 3;
    float2 a = red[(2 * y) * 8 + 2 * xq];
    float2 c = red[(2 * y) * 8 + 2 * xq + 1];
    float2 e = red[(2 * y + 1) * 8 + 2 * xq];
    float2 f = red[(2 * y + 1) * 8 + 2 * xq + 1];
    float2 r = make_float2(a.x + c.x + e.x + f.x, a.y + c.y + e.y + f.y);
    ws[80 + b * 16 + tid] = r;
    red[64 + tid] = r;
  }
  __syncthreads();

  if (tid < 4) {  // level-1: 2x2
    int y = tid >> 1, xq = tid & 1;
    float2 a = red[64 + (2 * y) * 4 + 2 * xq];
    float2 c = red[64 + (2 * y) * 4 + 2 * xq + 1];
    float2 e = red[64 + (2 * y + 1) * 4 + 2 * xq];
    float2 f = red[64 + (2 * y + 1) * 4 + 2 * xq + 1];
    float2 r = make_float2(a.x + c.x + e.x + f.x, a.y + c.y + e.y + f.y);
    ws[16 + b * 4 + tid] = r;
    red[80 + tid] = r;
  }
  __syncthreads();

  if (tid == 0) {  // level-0
    float2 a = red[80], c = red[81], e = red[82], f = red[83];
    ws[b] = make_float2(a.x + c.x + e.x + f.x, a.y + c.y + e.y + f.y);
  }
}

// ---------------------------------------------------------------------------
// Phase 3: per-pixel quadtree walk through the (L2-resident) stat pyramid.
// ---------------------------------------------------------------------------
__global__ __launch_bounds__(256) void qt_phase3(const float2* __restrict__ ws,
                                                 float* __restrict__ out) {
  const int gid = blockIdx.x * 256 + threadIdx.x;
  const int b = gid >> 18;
  const int p = gid & 262143;
  const int y = p >> 9, x = p & 511;

  float result = 0.0f;
#pragma unroll 1
  for (int d = 0; d <= 8; ++d) {
    const int sh = 9 - d;                 // log2(block side)
    const int by = y >> sh, bx = x >> sh;
    const float2 sq =
        ws[kOff[d] + ((size_t)b << (2 * d)) + ((size_t)by << d) + bx];
    const float n    = (float)(1 << (2 * sh));
    const float mean = sq.x / n;
    bool stop = true;
    if (d < 8) {
      float var = fmaxf((sq.y - sq.x * sq.x / n) / (n - 1.0f), 0.0f);
      stop = !(sqrtf(var) >= 0.2f);
    }
    if (stop) {
      result = mean;
      break;
    }
  }
  out[gid] = result;
}

// ---------------------------------------------------------------------------
extern "C" void kernel_launch(void* const* d_in, const int* in_sizes, int n_in,
                              void* d_out, int out_size, void* d_ws,
                              size_t ws_size, hipStream_t stream) {
  (void)in_sizes; (void)n_in; (void)out_size; (void)ws_size;
  const float* x = (const float*)d_in[0];
  float* out = (float*)d_out;
  float2* ws = (float2*)d_ws;  // needs 1,398,096 float2 = ~10.7 MB

  qt_phase1<<<16 * 64, 256, 0, stream>>>(x, ws);
  qt_phase2<<<16, 64, 0, stream>>>(ws);
  qt_phase3<<<16384, 256, 0, stream>>>(ws, out);
}